// Amplituedro_45183055954390
// MI455X (gfx1250) — compile-verified
//
#include <hip/hip_runtime.h>

typedef __attribute__((ext_vector_type(2))) float v2f;
typedef __attribute__((ext_vector_type(8))) float v8f;

#define B_TOTAL 65536
#define K_SEL   8
#define E_NUM   64
#define D_DIM   512
#define ESTRIDE 68    // padded K-stride (floats) of transposed vertex tile: conflict-free b64 loads
#define ASTRIDE 68    // padded agg row stride (floats): conflict-free, 8B-aligned b64 A loads
#define ROWS_PER_WAVE 16
#define WAVES_PER_BLOCK 16
#define THREADS_PER_BLOCK (WAVES_PER_BLOCK * 32)
#define ROWS_PER_BLOCK (ROWS_PER_WAVE * WAVES_PER_BLOCK)

__global__ void zero_eff_kernel(float* eff) {
  if (threadIdx.x == 0 && blockIdx.x == 0) *eff = 0.0f;
}

__global__ void __launch_bounds__(THREADS_PER_BLOCK)
moe_path_wmma_kernel(const int* __restrict__ eidx,
                     const float* __restrict__ ew,
                     const float* __restrict__ verts,
                     float* __restrict__ out,
                     float* __restrict__ eff)
{
  extern __shared__ float lds[];
  float* ldsVT = lds;                                              // [D_DIM][ESTRIDE] transposed
  float* ldsA  = ldsVT + D_DIM * ESTRIDE;                          // [16][16][ASTRIDE]
  float* ldsR  = ldsA + WAVES_PER_BLOCK * ROWS_PER_WAVE * ASTRIDE; // [16][16] recip totals

  const int tid  = threadIdx.x;
  const int wave = tid >> 5;
  const int lane = tid & 31;
  const int half = lane >> 4;   // 0: lanes 0-15, 1: lanes 16-31
  const int l16  = lane & 15;
  const int rowBase = blockIdx.x * ROWS_PER_BLOCK + wave * ROWS_PER_WAVE;

  // ---- Phase 1a: stage vertices (64x512 f32 = 128 KB) into LDS, TRANSPOSED ----
  // ldsVT[c][k] = verts[k][c]; stride 68 -> bank (4c + k) % 64, conflict-free b64 reads later.
  for (int i = tid; i < E_NUM * (D_DIM / 4); i += THREADS_PER_BLOCK) {
    const int r  = i >> 7;            // expert row, D_DIM/4 == 128
    const int c4 = i & 127;           // column quad
    float4 v = ((const float4*)(verts + r * D_DIM))[c4];
    ldsVT[(c4 * 4 + 0) * ESTRIDE + r] = v.x;
    ldsVT[(c4 * 4 + 1) * ESTRIDE + r] = v.y;
    ldsVT[(c4 * 4 + 2) * ESTRIDE + r] = v.z;
    ldsVT[(c4 * 4 + 3) * ESTRIDE + r] = v.w;
  }

  // ---- Phase 1b: build per-wave agg tile: 16 rows x 64 experts (sparse scatter) ----
  float* aggW = ldsA + wave * (ROWS_PER_WAVE * ASTRIDE);
  for (int i = lane; i < ROWS_PER_WAVE * ASTRIDE; i += 32) aggW[i] = 0.0f;
  if (lane < 16) {
    const int r = rowBase + lane;
    const int*   ip = eidx + r * K_SEL;
    const float* wp = ew   + r * K_SEL;
    float* aggRow = aggW + lane * ASTRIDE;
    float total = 0.0f;
    #pragma unroll
    for (int k = 0; k < K_SEL; ++k) {
      const int   e = ip[k];
      const float w = wp[k];
      if ((unsigned)e < (unsigned)E_NUM) { aggRow[e] += w; total += w; }
    }
    ldsR[wave * 16 + lane] = (total > 0.0f) ? (1.0f / total) : 1.0f;
  }
  __syncthreads();

  // ---- Phase 2: hoist A fragments (16 chunks of 16x4 f32) + per-row reciprocals ----
  // A layout (16x4 f32): VGPR0 = K=(0|2), VGPR1 = K=(1|3) for lanes (0-15 | 16-31), M = l16.
  const int koff = half ? 2 : 0;
  const float* aBase = aggW + l16 * ASTRIDE + koff;   // 8B-aligned (272*l16 + {0,8})
  v2f a[16];
  #pragma unroll
  for (int kc = 0; kc < 16; ++kc)
    a[kc] = *(const v2f*)(aBase + 4 * kc);            // ds_load_b64, small imm offsets
  float rt[8];
  #pragma unroll
  for (int j = 0; j < 8; ++j) rt[j] = ldsR[wave * 16 + half * 8 + j];

  float sq[8];
  #pragma unroll
  for (int j = 0; j < 8; ++j) sq[j] = 0.0f;

  // ---- Phase 3: 32 D-tiles of 16 cols; K=64 via 16x V_WMMA_F32_16X16X4_F32 ----
  // B layout (4x16 f32): VGPR0 = K=(0|2), VGPR1 = K=(1|3), N = l16.
  // Transposed LDS: all 16 chunk loads share one base, imm offsets 0..240 -> ds_load_b64.
  for (int dt = 0; dt < (D_DIM / 16); ++dt) {
    const int col = dt * 16 + l16;
    const float* vBase = ldsVT + col * ESTRIDE + koff; // 8B-aligned (272*col + {0,8})
    v8f c = {};
    #pragma unroll
    for (int kc = 0; kc < 16; ++kc) {
      v2f b = *(const v2f*)(vBase + 4 * kc);
      // 8 args: (neg_a, A, neg_b, B, c_mod, C, reuse_a, reuse_b)
      c = __builtin_amdgcn_wmma_f32_16x16x4_f32(false, a[kc], false, b,
                                                (short)0, c, false, false);
    }
    // C/D layout: VGPR j -> M = j + half*8, N = l16. Scale, write NT, track sumsq.
    #pragma unroll
    for (int j = 0; j < 8; ++j) {
      const float v = c[j] * rt[j];
      sq[j] += v * v;
      __builtin_nontemporal_store(v, out + (rowBase + half * 8 + j) * D_DIM + col);
    }
  }

  // ---- Phase 4: per-row L2 norms -> mean accumulator ----
  float acc = 0.0f;
  #pragma unroll
  for (int j = 0; j < 8; ++j) {
    float s = sq[j];
    s += __shfl_xor(s, 1, 16);
    s += __shfl_xor(s, 2, 16);
    s += __shfl_xor(s, 4, 16);
    s += __shfl_xor(s, 8, 16);
    acc += sqrtf(s);
  }
  if (l16 == 0) atomicAdd(eff, acc * (1.0f / (float)B_TOTAL));
}

extern "C" void kernel_launch(void* const* d_in, const int* in_sizes, int n_in,
                              void* d_out, int out_size, void* d_ws, size_t ws_size,
                              hipStream_t stream) {
  const int*   eidx  = (const int*)d_in[0];   // expert_indices [B,K] int32
  const float* ew    = (const float*)d_in[1]; // expert_weights [B,K] f32
  const float* verts = (const float*)d_in[2]; // vertices [E,D] f32
  float* out = (float*)d_out;                            // path [B,D] f32
  float* eff = out + (size_t)B_TOTAL * D_DIM;            // efficiency scalar

  zero_eff_kernel<<<1, 32, 0, stream>>>(eff);

  const size_t ldsBytes = (size_t)(D_DIM * ESTRIDE
                        + WAVES_PER_BLOCK * ROWS_PER_WAVE * ASTRIDE
                        + WAVES_PER_BLOCK * ROWS_PER_WAVE) * sizeof(float);
  moe_path_wmma_kernel<<<B_TOTAL / ROWS_PER_BLOCK, THREADS_PER_BLOCK, ldsBytes, stream>>>(
      eidx, ew, verts, out, eff);
}